// YoloLoss_54443005444658
// MI455X (gfx1250) — compile-verified
//
#include <hip/hip_runtime.h>
#include <hip/hip_bf16.h>
#include <math.h>

// YOLO loss, B=32 A=3 S=52 C=80.
// predictions: (B,A,S,S,85) f32, target: (B,A,S,S,6) f32, out: 4 f32 scalars.

typedef __attribute__((ext_vector_type(2))) float v2f;
typedef __attribute__((ext_vector_type(8))) float v8f;

#define NCELL (32 * 3 * 52 * 52)   // 259,584
#define PRED_C 85
#define TGT_C 6
#define NBLK 512                   // multiple of 4 for the WMMA reducer
#define NTHR 256

__device__ __forceinline__ float sigmoid_f(float x) {
  return 1.0f / (1.0f + __expf(-x));
}

__global__ void __launch_bounds__(NTHR)
yolo_partial(const float* __restrict__ pred,
             const float* __restrict__ tgt,
             float* __restrict__ partial /* [NBLK][16] */) {
  float a_noobj_bce = 0.0f, a_n_noobj = 0.0f;
  float a_obj_sq = 0.0f, a_n_obj = 0.0f;
  float a_box = 0.0f, a_cls = 0.0f;

  const int stride = gridDim.x * blockDim.x;
  for (int i = blockIdx.x * blockDim.x + threadIdx.x; i < NCELL; i += stride) {
    const float* p = pred + (size_t)i * PRED_C;
    const float* t = tgt + (size_t)i * TGT_C;

    // Prefetch next grid-stride cell (emits global_prefetch_b8 on gfx1250).
    __builtin_prefetch(pred + (size_t)(i + stride) * PRED_C, 0, 0);

    const float tobj = t[0];
    const bool obj = (tobj == 1.0f);
    const bool noobj = (tobj == 0.0f);

    // no-object BCE on channel 0
    const float x = p[0];
    const float bce = fmaxf(x, 0.0f) - x * tobj +
                      __logf(1.0f + __expf(-fabsf(x)));
    if (noobj) { a_noobj_bce += bce; a_n_noobj += 1.0f; }

    // midpoint IoU: pred[0:4] vs target[1:5]
    const float b1x = p[0], b1y = p[1], b1w = p[2], b1h = p[3];
    const float b2x = t[1], b2y = t[2], b2w = t[3], b2h = t[4];
    const float b1x1 = b1x - 0.5f * b1w, b1y1 = b1y - 0.5f * b1h;
    const float b1x2 = b1x + 0.5f * b1w, b1y2 = b1y + 0.5f * b1h;
    const float b2x1 = b2x - 0.5f * b2w, b2y1 = b2y - 0.5f * b2h;
    const float b2x2 = b2x + 0.5f * b2w, b2y2 = b2y + 0.5f * b2h;
    const float ix = fmaxf(fminf(b1x2, b2x2) - fmaxf(b1x1, b2x1), 0.0f);
    const float iy = fmaxf(fminf(b1y2, b2y2) - fmaxf(b1y1, b2y1), 0.0f);
    const float inter = ix * iy;
    const float ar1 = fabsf(b1w * b1h);
    const float ar2 = fabsf(b2w * b2h);
    const float iou = inter / (ar1 + ar2 - inter + 1e-6f);

    // object MSE
    const float p_obj = sigmoid_f(x);
    if (obj) {
      const float d = p_obj - iou * tobj;
      a_obj_sq += d * d;
      a_n_obj += 1.0f;
    }

    // box SE: sigmoid(pred[1:5]) vs target[1:5]
    float se = 0.0f;
#pragma unroll
    for (int j = 1; j <= 4; ++j) {
      const float d = sigmoid_f(p[j]) - t[j];
      se += d * d;
    }
    if (obj) a_box += se;

    // class NLL over 80 logits (pred[5:85])
    const float* lg = p + 5;
    float m = lg[0];
#pragma unroll 4
    for (int j = 1; j < 80; ++j) m = fmaxf(m, lg[j]);
    float s = 0.0f;
#pragma unroll 4
    for (int j = 0; j < 80; ++j) s += __expf(lg[j] - m);
    const int label = (int)t[5];
    const float nll = (m + __logf(s)) - lg[label];
    if (obj) a_cls += nll;
  }

  // Block reduction: 6 accumulators per thread.
  __shared__ float sh[NTHR * 6];
  const int tid = threadIdx.x;
  sh[tid * 6 + 0] = a_noobj_bce;
  sh[tid * 6 + 1] = a_n_noobj;
  sh[tid * 6 + 2] = a_obj_sq;
  sh[tid * 6 + 3] = a_n_obj;
  sh[tid * 6 + 4] = a_box;
  sh[tid * 6 + 5] = a_cls;
  __syncthreads();
  for (int off = NTHR / 2; off > 0; off >>= 1) {
    if (tid < off) {
#pragma unroll
      for (int j = 0; j < 6; ++j)
        sh[tid * 6 + j] += sh[(tid + off) * 6 + j];
    }
    __syncthreads();
  }
  // Write a full 16-wide row (pad with zeros) so the WMMA reducer reads
  // deterministic data (d_ws is poisoned, so every column must be written).
  if (tid < 16) {
    partial[blockIdx.x * 16 + tid] = (tid < 6) ? sh[tid] : 0.0f;
  }
}

// One wave32 reduces partial[NBLK][16] column-wise with V_WMMA_F32_16X16X4_F32:
// A = ones(16x4) so D[m][n] = sum_k B[k][n] + C[m][n]. Exact f32 reduction.
__global__ void __launch_bounds__(32)
yolo_final(const float* __restrict__ partial, float* __restrict__ out) {
  const int lane = threadIdx.x;         // 0..31, EXEC all-1s for WMMA
  const int col = lane & 15;            // N index
  const int kb = (lane >> 4) * 2;       // lanes 0-15 hold K=0,1; 16-31 hold K=2,3

  v2f a; a.x = 1.0f; a.y = 1.0f;        // ones A-matrix (16x4 f32)
  v8f c = {};

  for (int k0 = 0; k0 < NBLK; k0 += 4) {
    v2f b;
    b.x = partial[(size_t)(k0 + kb + 0) * 16 + col];
    b.y = partial[(size_t)(k0 + kb + 1) * 16 + col];
    c = __builtin_amdgcn_wmma_f32_16x16x4_f32(
        /*neg_a=*/false, a, /*neg_b=*/false, b,
        /*c_mod=*/(short)0, c, /*reuse_a=*/false, /*reuse_b=*/false);
  }

  // Row M=0 of D (VGPR 0, lanes 0-15) holds the 16 column sums.
  const float colsum = c[0];
  const float s_noobj_bce = __shfl(colsum, 0, 32);
  const float s_n_noobj   = __shfl(colsum, 1, 32);
  const float s_obj_sq    = __shfl(colsum, 2, 32);
  const float s_n_obj     = __shfl(colsum, 3, 32);
  const float s_box       = __shfl(colsum, 4, 32);
  const float s_cls       = __shfl(colsum, 5, 32);

  if (lane == 0) {
    // Reference return order: (5*box, 1*object, 0.5*no_object, 1*class)
    out[0] = 5.0f * s_box / fmaxf(s_n_obj * 4.0f, 1.0f);
    out[1] = 1.0f * s_obj_sq / fmaxf(s_n_obj, 1.0f);
    out[2] = 0.5f * s_noobj_bce / fmaxf(s_n_noobj, 1.0f);
    out[3] = 1.0f * s_cls / fmaxf(s_n_obj, 1.0f);
  }
}

extern "C" void kernel_launch(void* const* d_in, const int* in_sizes, int n_in,
                              void* d_out, int out_size, void* d_ws, size_t ws_size,
                              hipStream_t stream) {
  const float* pred = (const float*)d_in[0];   // (B,A,S,S,85) f32
  const float* tgt  = (const float*)d_in[1];   // (B,A,S,S,6)  f32
  float* P   = (float*)d_ws;                   // [NBLK][16] partials (32 KB)
  float* out = (float*)d_out;                  // 4 f32 scalars

  yolo_partial<<<NBLK, NTHR, 0, stream>>>(pred, tgt, P);
  yolo_final<<<1, 32, 0, stream>>>(P, out);
}